// GraphConvolution_21569325760885
// MI455X (gfx1250) — compile-verified
//
#include <hip/hip_runtime.h>

typedef __attribute__((ext_vector_type(2))) float v2f;
typedef __attribute__((ext_vector_type(8))) float v8f;

#define D 64

// ---------------------------------------------------------------------------
// init: out[i*64+c] = bias[c]  (folds the "+ bias" into initialization),
//       deg[i] = 0
// ---------------------------------------------------------------------------
__global__ void gcn_init_kernel(float* __restrict__ out,
                                const float* __restrict__ bias,
                                unsigned* __restrict__ deg,
                                int n_out, int n_nodes) {
    int idx = blockIdx.x * blockDim.x + threadIdx.x;
    if (idx < n_out) out[idx] = bias[idx & (D - 1)];
    if (idx < n_nodes) deg[idx] = 0u;
}

// ---------------------------------------------------------------------------
// degree: deg[row[e]] += 1  (native u32 atomics)
// ---------------------------------------------------------------------------
__global__ void gcn_degree_kernel(const long long* __restrict__ row,
                                  unsigned* __restrict__ deg, int E) {
    int e = blockIdx.x * blockDim.x + threadIdx.x;
    if (e < E) atomicAdd(&deg[(int)row[e]], 1u);
}

// ---------------------------------------------------------------------------
// dis[i] = deg[i] > 0 ? deg[i]^-0.5 : 0
// ---------------------------------------------------------------------------
__global__ void gcn_rsqrt_kernel(const unsigned* __restrict__ deg,
                                 float* __restrict__ dis, int n) {
    int i = blockIdx.x * blockDim.x + threadIdx.x;
    if (i < n) {
        unsigned d = deg[i];
        dis[i] = (d > 0u) ? rsqrtf((float)d) : 0.0f;
    }
}

// ---------------------------------------------------------------------------
// xt = x @ W using V_WMMA_F32_16X16X4_F32.
// One wave per 16x16 output tile; block of 128 threads = 4 waves covering
// the 64 output columns; gridDim.x = N/16 row tiles (100000 % 16 == 0).
//
// ISA layouts (wave32):
//   A 16x4 f32 : lane 0-15 -> M=lane, lanes 16-31 -> M=lane-16;
//                VGPR v holds K = v + (lane>>4)*2
//   B 4x16 f32 : lane -> N = lane&15; VGPR v holds K = v + (lane>>4)*2
//   C/D 16x16  : VGPR r, N = lane&15, M = r + (lane>>4)*8
// ---------------------------------------------------------------------------
__global__ void gcn_gemm_wmma_kernel(const float* __restrict__ x,
                                     const float* __restrict__ w,
                                     float* __restrict__ xt) {
    const int lane = threadIdx.x & 31;
    const int wave = threadIdx.x >> 5;        // 0..3 -> column tile
    const int m0   = blockIdx.x * 16;
    const int n0   = wave * 16;

    const int l15  = lane & 15;
    const int lh   = lane >> 4;               // lane-half (0/1)
    const int mrow = m0 + l15;                // A row for this lane
    const int ncol = n0 + l15;                // B/D column for this lane

    v8f acc = {};
#pragma unroll
    for (int k = 0; k < D; k += 4) {
        v2f a, b;
        const int ka = k + lh * 2;
        a.x = x[mrow * D + ka + 0];
        a.y = x[mrow * D + ka + 1];
        b.x = w[(ka + 0) * D + ncol];
        b.y = w[(ka + 1) * D + ncol];
        // 8 args: (neg_a, A, neg_b, B, c_mod, C, reuse_a, reuse_b)
        acc = __builtin_amdgcn_wmma_f32_16x16x4_f32(
            false, a, false, b, (short)0, acc, false, false);
    }

#pragma unroll
    for (int r = 0; r < 8; ++r) {
        const int m = m0 + r + lh * 8;
        xt[m * D + ncol] = acc[r];
    }
}

// ---------------------------------------------------------------------------
// Edge aggregation: one wave per edge (grid-stride over edges).
//   norm = dis[row]*dis[col]
//   out[col, :] += norm * xt[row, :]
// Each lane handles columns {lane, lane+32}: two coalesced 128B gathers and
// two native fp32 atomic adds per lane. xt/out live in L2 (25.6 MB each).
// ---------------------------------------------------------------------------
__global__ void gcn_aggregate_kernel(const long long* __restrict__ row,
                                     const long long* __restrict__ col,
                                     const float* __restrict__ dis,
                                     const float* __restrict__ xt,
                                     float* __restrict__ out, int E) {
    const int lane  = threadIdx.x & 31;
    int w           = (blockIdx.x * blockDim.x + threadIdx.x) >> 5;
    const int nw    = (gridDim.x * blockDim.x) >> 5;

    for (int e = w; e < E; e += nw) {
        const int r = (int)row[e];           // wave-uniform
        const int c = (int)col[e];
        const float norm = dis[r] * dis[c];
        const float v0 = xt[r * D + lane];
        const float v1 = xt[r * D + 32 + lane];
        unsafeAtomicAdd(&out[c * D + lane],      norm * v0);
        unsafeAtomicAdd(&out[c * D + 32 + lane], norm * v1);
    }
}

// ---------------------------------------------------------------------------
extern "C" void kernel_launch(void* const* d_in, const int* in_sizes, int n_in,
                              void* d_out, int out_size, void* d_ws, size_t ws_size,
                              hipStream_t stream) {
    const float*     x    = (const float*)d_in[0];          // [N, 64]
    const long long* eidx = (const long long*)d_in[1];      // [2, E] int64
    const float*     wgt  = (const float*)d_in[2];          // [64, 64]
    const float*     bias = (const float*)d_in[3];          // [64]
    float*           out  = (float*)d_out;                  // [N, 64]

    const int n_nodes = in_sizes[0] / D;
    const int E       = in_sizes[1] / 2;

    const long long* row = eidx;         // edge_index[0]
    const long long* col = eidx + E;     // edge_index[1]

    // workspace layout
    float*    xt  = (float*)d_ws;                                 // N*64 f32
    unsigned* deg = (unsigned*)((char*)d_ws +
                                (size_t)n_nodes * D * sizeof(float));
    float*    dis = (float*)(deg + n_nodes);

    // 1) init out = bias, deg = 0
    {
        int n = n_nodes * D;
        int blocks = (n + 255) / 256;
        gcn_init_kernel<<<blocks, 256, 0, stream>>>(out, bias, deg, n, n_nodes);
    }
    // 2) degree count
    gcn_degree_kernel<<<(E + 255) / 256, 256, 0, stream>>>(row, deg, E);
    // 3) deg^-0.5
    gcn_rsqrt_kernel<<<(n_nodes + 255) / 256, 256, 0, stream>>>(deg, dis, n_nodes);
    // 4) xt = x @ W  (WMMA f32 16x16x4)
    gcn_gemm_wmma_kernel<<<n_nodes / 16, 128, 0, stream>>>(x, wgt, xt);
    // 5) scatter-add aggregation
    gcn_aggregate_kernel<<<2048, 256, 0, stream>>>(row, col, dis, xt, out, E);
}